// GCN_8504035246475
// MI455X (gfx1250) — compile-verified
//
#include <hip/hip_runtime.h>

typedef __attribute__((ext_vector_type(2))) float v2f;
typedef __attribute__((ext_vector_type(8))) float v8f;

// ---------------- degree / norm precompute ----------------

__global__ void deg_init_kernel(float* __restrict__ deg, int N) {
    int i = blockIdx.x * blockDim.x + threadIdx.x;
    if (i < N) deg[i] = 1.0f;   // self-loop contributes 1 to every node
}

__global__ void deg_count_kernel(const int* __restrict__ dst, float* __restrict__ deg, int E) {
    int e = blockIdx.x * blockDim.x + threadIdx.x;
    if (e < E) atomicAdd(&deg[dst[e]], 1.0f);
}

__global__ void dinv_kernel(float* __restrict__ deg, int N) {
    int i = blockIdx.x * blockDim.x + threadIdx.x;
    if (i < N) deg[i] = rsqrtf(deg[i]);   // deg >= 1 always (self-loops)
}

__global__ void norm_kernel(const int* __restrict__ src, const int* __restrict__ dst,
                            const float* __restrict__ dinv, float* __restrict__ norm, int E) {
    int e = blockIdx.x * blockDim.x + threadIdx.x;
    if (e < E) norm[e] = dinv[src[e]] * dinv[dst[e]];
}

// ---------------- WMMA GEMM: H[N,64] = X[N,64] @ W[64,64] ----------------
// One wave computes a 16x64 output tile via V_WMMA_F32_16X16X4_F32.
// 8 waves per block -> 128 rows per block. W cached in LDS (16 KB).

__global__ __launch_bounds__(256) void gemm_wmma_kernel(const float* __restrict__ X,
                                                        const float* __restrict__ W,
                                                        float* __restrict__ H, int N) {
    __shared__ float Ws[64 * 64];
    const int tid = threadIdx.x;

    // cooperative load of W into LDS (4096 floats = 1024 float4)
    for (int i = tid; i < 1024; i += 256) {
        ((float4*)Ws)[i] = ((const float4*)W)[i];
    }
    __syncthreads();

    const int wave = tid >> 5;          // 0..7
    const int lane = tid & 31;
    const int lm   = lane & 15;         // M (for A) / N (for B,C) within tile
    const int kh   = lane >> 4;         // K half-select (0 or 1)
    const int m0   = blockIdx.x * 128 + wave * 16;

    // A-fragment source row (clamped so EXEC stays all-1s through the WMMAs)
    int row = m0 + lm;
    if (row >= N) row = N - 1;
    const float* xrow = X + (size_t)row * 64;

    // Preload all 16 A fragments (K = 0..63 in steps of 4).
    // Layout (32b A 16x4): lanes 0-15 -> K = k0+0/k0+1, lanes 16-31 -> K = k0+2/k0+3.
    v2f a[16];
#pragma unroll
    for (int ks = 0; ks < 16; ++ks) {
        const int k = ks * 4 + 2 * kh;
        a[ks] = *(const v2f*)(xrow + k);   // 8B-aligned: k is even
    }

#pragma unroll
    for (int nt = 0; nt < 4; ++nt) {
        const int n = nt * 16 + lm;
        v8f c = {};
#pragma unroll
        for (int ks = 0; ks < 16; ++ks) {
            const int k = ks * 4 + 2 * kh;
            v2f b;
            b.x = Ws[k * 64 + n];
            b.y = Ws[(k + 1) * 64 + n];
            c = __builtin_amdgcn_wmma_f32_16x16x4_f32(
                    /*neg_a=*/false, a[ks], /*neg_b=*/false, b,
                    /*c_mod=*/(short)0, c, /*reuse_a=*/false, /*reuse_b=*/false);
        }
        // C/D layout: lanes 0-15: M = v ; lanes 16-31: M = v + 8 ; N = lm
#pragma unroll
        for (int v = 0; v < 8; ++v) {
            const int m = (lane < 16) ? v : (v + 8);
            const int r = m0 + m;
            if (r < N) H[(size_t)r * 64 + nt * 16 + lm] = c[v];
        }
    }
}

// ---------------- aggregation ----------------

// out[i,d] = h[i,d]*dinv[i]^2 + b[d]   (self-loop message + bias)
__global__ void agg_init_kernel(const float* __restrict__ h, const float* __restrict__ dinv,
                                const float* __restrict__ bias, float* __restrict__ out, int N) {
    int idx = blockIdx.x * blockDim.x + threadIdx.x;
    if (idx < N * 64) {
        const int i = idx >> 6;
        const int d = idx & 63;
        const float di = dinv[i];
        out[idx] = h[idx] * di * di + bias[d];
    }
}

// out[dst,d] += h[src,d] * norm[e]  over all edges
__global__ void scatter_kernel(const int* __restrict__ src, const int* __restrict__ dst,
                               const float* __restrict__ h, const float* __restrict__ norm,
                               float* __restrict__ out, int E) {
    int idx = blockIdx.x * blockDim.x + threadIdx.x;
    if (idx < E * 64) {
        const int e = idx >> 6;
        const int d = idx & 63;
        const int s = src[e];
        const int t = dst[e];
        atomicAdd(&out[t * 64 + d], h[s * 64 + d] * norm[e]);
    }
}

// ---------------- launcher ----------------

extern "C" void kernel_launch(void* const* d_in, const int* in_sizes, int n_in,
                              void* d_out, int out_size, void* d_ws, size_t ws_size,
                              hipStream_t stream) {
    const float* x  = (const float*)d_in[0];
    const int*   ei = (const int*)d_in[1];
    const float* W[3] = { (const float*)d_in[2], (const float*)d_in[4], (const float*)d_in[6] };
    const float* b[3] = { (const float*)d_in[3], (const float*)d_in[5], (const float*)d_in[7] };

    const int N = in_sizes[0] / 64;
    const int E = in_sizes[1] / 2;
    const int* src = ei;          // edge_index[0]
    const int* dst = ei + E;      // edge_index[1]

    // workspace layout (floats): dinv[N] | norm[E] | h[N*64] | bufA[N*64]
    float* ws   = (float*)d_ws;
    float* dinv = ws;
    float* norm = dinv + N;
    float* h    = norm + E;
    float* bufA = h + (size_t)N * 64;
    float* out  = (float*)d_out;

    const int T = 256;
    // --- degree / normalization (same for every layer) ---
    deg_init_kernel<<<(N + T - 1) / T, T, 0, stream>>>(dinv, N);
    deg_count_kernel<<<(E + T - 1) / T, T, 0, stream>>>(dst, dinv, E);
    dinv_kernel<<<(N + T - 1) / T, T, 0, stream>>>(dinv, N);
    norm_kernel<<<(E + T - 1) / T, T, 0, stream>>>(src, dst, dinv, norm, E);

    // --- 3 GCN layers ---
    const int gemmBlocks = (N + 127) / 128;
    const int nodeElems  = N * 64;
    const long long edgeElems = (long long)E * 64;
    const int edgeBlocks = (int)((edgeElems + T - 1) / T);

    const float* cur = x;
    for (int l = 0; l < 3; ++l) {
        gemm_wmma_kernel<<<gemmBlocks, T, 0, stream>>>(cur, W[l], h, N);
        float* tgt = (l == 2) ? out : bufA;   // bufA safely reusable: gemm consumed it into h
        agg_init_kernel<<<(nodeElems + T - 1) / T, T, 0, stream>>>(h, dinv, b[l], tgt, N);
        scatter_kernel<<<edgeBlocks, T, 0, stream>>>(src, dst, h, norm, tgt, (int)E);
        cur = tgt;
    }
}